// Loss_39883066311212
// MI455X (gfx1250) — compile-verified
//
#include <hip/hip_runtime.h>

// ---- problem constants (from reference) ----
#define S7     7
#define ECH    30                    // E = B*5 + C = 30 channels per cell
#define CELLS  (16384 * S7 * S7)     // 802816 cells
#define TPB    256
#define NBLK   (CELLS / TPB)         // 3136 (exact: 802816 = 3136*256)
#define TILE_FLOATS (TPB * ECH)      // 7680 floats per tensor tile (30720 B)
#define TILE_V4     (TILE_FLOATS / 4)// 1920 float4 loads per tensor tile

typedef float    v2f   __attribute__((ext_vector_type(2)));
typedef float    v4f   __attribute__((ext_vector_type(4)));
typedef float    v8f   __attribute__((ext_vector_type(8)));
typedef unsigned u32x4 __attribute__((ext_vector_type(4)));
typedef int      i32x4 __attribute__((ext_vector_type(4)));
typedef int      i32x8 __attribute__((ext_vector_type(8)));

// ---------- TDM path selection ----------
#if defined(__HIP_DEVICE_COMPILE__) && __has_builtin(__builtin_amdgcn_tensor_load_to_lds)
  #define USE_TDM 1
  #if __has_include(<hip/amd_detail/amd_gfx1250_TDM.h>)
    #define TDM_6ARG 1   // amdgpu-toolchain (clang-23 / therock headers): 6-arg form
  #else
    #define TDM_6ARG 0   // ROCm 7.2 (clang-22): 5-arg form
  #endif
#else
  #define USE_TDM 0
#endif

#if USE_TDM
// DMA a contiguous run of `nelem` f32 (nelem*4 bytes, nelem <= 65535) from
// global memory into LDS via TENSOR_LOAD_TO_LDS.  D# built per ISA ch.8:
//   group0: count=1 | lds_addr[63:32] | global_addr[120:64] | type=2 (bits 127:126)
//   group1: data_size=4B (bits 17:16), tensor_dim0=nelem (79:48),
//           tensor_dim1=1 (111:80), tile_dim0=nelem (127:112),
//           tile_dim1=tile_dim2=0 (1D tile), tensor_dim0_stride=nelem (207:160)
//   groups 2/3: zero (dims 2+ unused)
__device__ __forceinline__ void tdm_load_f32_1d(const float* gsrc, float* ldsdst,
                                                unsigned nelem) {
    union { unsigned long long u64; unsigned u32[2]; } ga;
    ga.u64 = (unsigned long long)(uintptr_t)gsrc;
    // generic pointer to LDS: low 32 bits are the LDS byte offset
    const unsigned ldsoff = (unsigned)(uintptr_t)ldsdst;

    u32x4 g0;
    g0.x = 1u;                                      // count=1, user descriptor
    g0.y = ldsoff;                                  // lds_addr
    g0.z = ga.u32[0];                               // global_addr[31:0]
    g0.w = (ga.u32[1] & 0x01FFFFFFu) | 0x80000000u; // global_addr[56:32] | type=2

    i32x8 g1;
    g1[0] = 0x20000;                                // data_size=4B, mask=0, no flags
    g1[1] = (int)((nelem & 0xFFFFu) << 16);         // tensor_dim0[15:0]
    g1[2] = (int)((nelem >> 16) & 0xFFFFu)          // tensor_dim0[31:16]
          | (1 << 16);                              // tensor_dim1 = 1
    g1[3] = (int)((nelem & 0xFFFFu) << 16);         // tile_dim0 = nelem
    g1[4] = 0;                                      // tile_dim1=0, tile_dim2=0
    g1[5] = (int)nelem;                             // tensor_dim0_stride[31:0]
    g1[6] = 0;                                      // stride hi / dim1_stride lo
    g1[7] = 0;

    i32x4 gz = {0, 0, 0, 0};
#if TDM_6ARG
    i32x8 gz8 = {0, 0, 0, 0, 0, 0, 0, 0};
    __builtin_amdgcn_tensor_load_to_lds(g0, g1, gz, gz, gz8, 0);
#else
    __builtin_amdgcn_tensor_load_to_lds(g0, g1, gz, gz, 0);
#endif
}
#endif // USE_TDM

// Sum x across all 32 lanes of the wave using V_WMMA_F32_16X16X4_F32.
// A (16x4 f32) layout: lanes 0-15 supply K=0 (vgpr0) / K=1 (vgpr1),
// lanes 16-31 supply K=2 / K=3.  With A = (x, 0) per lane and B = ones
// (all-ones is layout invariant), D[m][n] = x[m] + x[m+16] for every n.
// Per-lane sum of the 8 D vgprs gives the half-wave sums (M=0..7 in lanes
// 0-15, M=8..15 in lanes 16-31); one xor-16 shuffle completes the total.
__device__ __forceinline__ float wave_sum_wmma(float x) {
    v2f a; a.x = x;    a.y = 0.0f;
    v2f b; b.x = 1.0f; b.y = 1.0f;
    v8f c = {};
    v8f d = __builtin_amdgcn_wmma_f32_16x16x4_f32(
        /*neg_a=*/false, a, /*neg_b=*/false, b,
        /*c_mod=*/(short)0, c, /*reuse_a=*/false, /*reuse_b=*/false);
    float s = d[0] + d[1] + d[2] + d[3] + d[4] + d[5] + d[6] + d[7];
    s += __shfl_xor(s, 16, 32);
    return s; // full 32-lane sum, broadcast to all lanes
}

__global__ __launch_bounds__(TPB) void yolo_loss_partial(
    const float* __restrict__ pred, const float* __restrict__ tgt,
    float* __restrict__ partial) {
    __shared__ float ldsP[TILE_FLOATS];
    __shared__ float ldsT[TILE_FLOATS];
    __shared__ float red[(TPB / 32) * 5];

    const int tid = threadIdx.x;
    const long long tileBase = (long long)blockIdx.x * TILE_FLOATS;

#if USE_TDM
    // ---- DMA both 30 KB tiles into LDS with the Tensor Data Mover ----
    if (tid == 0) {
        tdm_load_f32_1d(pred + tileBase, ldsP, TILE_FLOATS);
        tdm_load_f32_1d(tgt  + tileBase, ldsT, TILE_FLOATS);
        __builtin_amdgcn_s_wait_tensorcnt(0);
    }
    __syncthreads();
#else
    // ---- fallback: stage tiles through LDS with coalesced NT b128 loads ----
    {
        const v4f* gp = (const v4f*)(pred + tileBase);
        const v4f* gt = (const v4f*)(tgt + tileBase);
        v4f* lp = (v4f*)ldsP;
        v4f* lt = (v4f*)ldsT;
        for (int i = tid; i < TILE_V4; i += TPB) {
            lp[i] = __builtin_nontemporal_load(&gp[i]);
            lt[i] = __builtin_nontemporal_load(&gt[i]);
        }
    }
    __syncthreads();
#endif

    // ---- per-cell loss (one cell per lane; stride-30 LDS reads are
    //      bank-conflict-free across a wave32) ----
    const float* p = &ldsP[tid * ECH];
    const float* t = &ldsT[tid * ECH];

    const float t0 = t[0], t1 = t[1], t2 = t[2], t3 = t[3], t4 = t[4];
    const float m  = (t4 > 0.0f)  ? 1.0f : 0.0f;
    const float nm = (t4 == 0.0f) ? 1.0f : 0.0f;
    const float a2 = (t2 - t0) * (t3 - t1);

    float iou[2];
#pragma unroll
    for (int bb = 0; bb < 2; ++bb) {
        const float p0 = p[bb * 5 + 0], p1 = p[bb * 5 + 1];
        const float p2 = p[bb * 5 + 2], p3 = p[bb * 5 + 3];
        const float ltx = fmaxf(p0, t0), lty = fmaxf(p1, t1);
        const float rbx = fminf(p2, t2), rby = fminf(p3, t3);
        const float w = fmaxf(rbx - ltx, 0.0f);
        const float h = fmaxf(rby - lty, 0.0f);
        const float inter = w * h;
        const float a1 = (p2 - p0) * (p3 - p1);
        iou[bb] = inter / (a1 + a2 - inter);
    }
    const int   j       = (iou[1] > iou[0]) ? 1 : 0; // first-max on tie, like argmax
    const float max_iou = fmaxf(iou[0], iou[1]);

    const float* pj = p + j * 5;
    const float* tj = t + j * 5;

    const float dx = pj[0] - tj[0], dy = pj[1] - tj[1];
    const float sw = __builtin_sqrtf(pj[2]) - __builtin_sqrtf(tj[2]);
    const float sh = __builtin_sqrtf(pj[3]) - __builtin_sqrtf(tj[3]);
    const float dob = pj[4] - max_iou;
    const float d4 = p[4] - t4, d9 = p[9] - t[9];

    float cls = 0.0f;
#pragma unroll
    for (int k = 10; k < 30; ++k) {
        const float d = p[k] - t[k];
        cls += d * d;
    }

    float acc[5];
    acc[0] = m * (dx * dx + dy * dy);       // loss_xy
    acc[1] = m * (sw * sw + sh * sh);       // loss_wh
    acc[2] = m * (dob * dob);               // loss_obj
    acc[3] = nm * (d4 * d4 + d9 * d9);      // loss_noobj
    acc[4] = m * cls;                       // loss_class

    // ---- wave reduction via WMMA, then cross-wave via LDS ----
    const int wave = tid >> 5;
    const int lane = tid & 31;
#pragma unroll
    for (int k = 0; k < 5; ++k) {
        const float tot = wave_sum_wmma(acc[k]);
        if (lane == 0) red[wave * 5 + k] = tot;
    }
    __syncthreads();
    if (tid < 5) {
        float s = 0.0f;
#pragma unroll
        for (int w = 0; w < TPB / 32; ++w) s += red[w * 5 + tid];
        partial[(long long)blockIdx.x * 5 + tid] = s;
    }
}

__global__ __launch_bounds__(TPB) void yolo_loss_final(
    const float* __restrict__ partial, float* __restrict__ out, int nrows) {
    __shared__ float red[(TPB / 32) * 5];
    const int tid = threadIdx.x;

    float acc[5] = {0.0f, 0.0f, 0.0f, 0.0f, 0.0f};
    for (int r = tid; r < nrows; r += TPB) {
#pragma unroll
        for (int k = 0; k < 5; ++k) acc[k] += partial[r * 5 + k];
    }

    const int wave = tid >> 5;
    const int lane = tid & 31;
#pragma unroll
    for (int k = 0; k < 5; ++k) {
        const float tot = wave_sum_wmma(acc[k]);
        if (lane == 0) red[wave * 5 + k] = tot;
    }
    __syncthreads();
    if (tid < 5) {
        float s = 0.0f;
#pragma unroll
        for (int w = 0; w < TPB / 32; ++w) s += red[w * 5 + tid];
        out[tid] = s;
    }
}

extern "C" void kernel_launch(void* const* d_in, const int* in_sizes, int n_in,
                              void* d_out, int out_size, void* d_ws, size_t ws_size,
                              hipStream_t stream) {
    (void)in_sizes; (void)n_in; (void)out_size; (void)ws_size;
    const float* pred = (const float*)d_in[0];
    const float* tgt  = (const float*)d_in[1];
    float* out = (float*)d_out;
    float* ws  = (float*)d_ws;   // needs 3136*5*4 = 62720 bytes

    yolo_loss_partial<<<NBLK, TPB, 0, stream>>>(pred, tgt, ws);
    yolo_loss_final<<<1, TPB, 0, stream>>>(ws, out, NBLK);
}